// BPKDV3_24979529793865
// MI455X (gfx1250) — compile-verified
//
#include <hip/hip_runtime.h>
#include <stdint.h>

// ---------------------------------------------------------------------------
// BPKD edge/body channel-wise KD loss for MI455X (gfx1250).
// Memory-bound streaming reduction (~121 MB @ 23.3 TB/s => ~5.2us floor).
// CDNA5 path: Tensor Data Mover (tensor_load_to_lds) double-buffered
// global->LDS staging + s_wait_tensorcnt, wave32 shfl reductions.
// ---------------------------------------------------------------------------

#define BATCH 4
#define NCLS 14
#define HDIM 512
#define WDIM 512
#define HW (HDIM * WDIM)                   // 262144
#define CHUNKS 128
#define CHUNK_ELEMS (HW / CHUNKS)          // 2048 floats = 8KB per tensor
#define TPB 256                            // 8 wave32 per workgroup
#define EPT (CHUNK_ELEMS / TPB)            // 8 elements per thread
#define NCH_USED (BATCH * (NCLS - 1))      // 52 (class 0 contributes exactly 0)

typedef unsigned int u32x4 __attribute__((ext_vector_type(4)));
typedef int          i32x4 __attribute__((ext_vector_type(4)));
typedef int          i32x8 __attribute__((ext_vector_type(8)));

#ifndef __has_builtin
#define __has_builtin(x) 0
#endif

#if __has_builtin(__builtin_amdgcn_tensor_load_to_lds) && \
    __has_builtin(__builtin_amdgcn_s_wait_tensorcnt)
#define USE_TDM 1
#else
#define USE_TDM 0
#endif

// The AMDGPU builtins are invisible to the x86 host pass; only require them
// on the device pass so a silent scalar fallback can't hide.
#if defined(__HIP_DEVICE_COMPILE__) && !USE_TDM
#error "gfx1250 TDM builtins not available on device pass"
#endif

#if USE_TDM
// Issue one TDM DMA: contiguous row of CHUNK_ELEMS f32 from global -> LDS.
// D# group0: count=1, lds_addr, 57-bit global addr, type=2 ("image").
// D# group1: data_size=4B (code 2), tile = CHUNK_ELEMS x 1, tensor_dim0 and
// stride = 2^30 (no OOB clipping), workgroup_mask=0 (not in a cluster).
__device__ __forceinline__ void tdm_load_row(const float* gsrc, float* lds_dst) {
    unsigned long long ga = (unsigned long long)(uintptr_t)gsrc;
    u32x4 g0;
    g0[0] = 1u;                                            // count=1, user mode
    g0[1] = (unsigned)(uintptr_t)lds_dst;                  // LDS byte offset (flat low 32b)
    g0[2] = (unsigned)ga;                                  // global_addr[31:0]
    g0[3] = (unsigned)((ga >> 32) & 0x1FFFFFFull) | (2u << 30); // addr[56:32] | type=2
    i32x8 g1;
    g1[0] = 0x00020000;                                    // data_size = 4 bytes
    g1[1] = 0;                                             // tensor_dim0[15:0]=0 (dim0=1<<30)
    g1[2] = 0x00014000;                                    // dim0[31:16]=0x4000, dim1=1
    g1[3] = (int)((unsigned)CHUNK_ELEMS << 16);            // tile_dim0 = 2048
    g1[4] = 1;                                             // tile_dim1 = 1, tile_dim2 = 0
    g1[5] = 0x40000000;                                    // tensor_dim0_stride = 1<<30
    g1[6] = 0;
    g1[7] = 0;
    i32x4 z4 = {0, 0, 0, 0};
#if __clang_major__ >= 23
    i32x8 z8 = {0, 0, 0, 0, 0, 0, 0, 0};
    __builtin_amdgcn_tensor_load_to_lds(g0, g1, z4, z4, z8, 0);
#else
    __builtin_amdgcn_tensor_load_to_lds(g0, g1, z4, z4, 0);
#endif
}
#endif // USE_TDM

// ---------------------------------------------------------------------------
// Main kernel (placed first so the disasm snippet shows the TDM path):
// per (batch, chunk) block, loop classes 1..13; TDM-stage S/T tiles into LDS
// (double-buffered, wave0 issues), accumulate 3 streaming-softmax sums per
// mask type, wave32 shfl tree + LDS reduce, write per-block partials.
// ---------------------------------------------------------------------------
__global__ __launch_bounds__(TPB) void bpkd_main_kernel(
        const float* __restrict__ S, const float* __restrict__ T,
        const unsigned* __restrict__ maskbuf, float* __restrict__ partials) {
#if USE_TDM
    __shared__ __align__(16) float sbuf[2][CHUNK_ELEMS];
    __shared__ __align__(16) float tbuf[2][CHUNK_ELEMS];
#endif
    __shared__ float red[TPB / 32][6];

    const int b = blockIdx.x / CHUNKS;
    const int chunk = blockIdx.x % CHUNKS;
    const int tid = threadIdx.x;

    // Packed edge/body masks: 8 pixels per thread, register-resident across
    // the whole 13-class loop (gt-derived data read once per chunk).
    const uint4* mbase = (const uint4*)(maskbuf + (size_t)b * HW +
                                        (size_t)chunk * CHUNK_ELEMS);
    const uint4 mk0 = mbase[tid];
    const uint4 mk1 = mbase[tid + TPB];
    const unsigned pm[EPT] = {mk0.x, mk0.y, mk0.z, mk0.w,
                              mk1.x, mk1.y, mk1.z, mk1.w};

    auto chan_ptr = [&](const float* base, int c) -> const float* {
        return base + ((size_t)(b * NCLS + c)) * HW + (size_t)chunk * CHUNK_ELEMS;
    };

#if USE_TDM
    // TDM ignores EXEC, so gate with an SGPR-uniform branch: wave 0 only.
    const bool issuer = (__builtin_amdgcn_readfirstlane((int)(tid >> 5)) == 0);
    if (issuer) {
        tdm_load_row(chan_ptr(S, 1), &sbuf[0][0]);
        tdm_load_row(chan_ptr(T, 1), &tbuf[0][0]);
    }
#endif

    for (int c = 1; c < NCLS; ++c) {
        const int cur = (c - 1) & 1;
#if USE_TDM
        if (issuer) {
            if (c + 1 < NCLS) {  // prefetch next channel into the other buffer
                tdm_load_row(chan_ptr(S, c + 1), &sbuf[c & 1][0]);
                tdm_load_row(chan_ptr(T, c + 1), &tbuf[c & 1][0]);
                __builtin_amdgcn_s_wait_tensorcnt(2);  // current pair complete
            } else {
                __builtin_amdgcn_s_wait_tensorcnt(0);
            }
        }
        __syncthreads();  // release all waves once tile for class c is in LDS
        const float4 sv0 = ((const float4*)&sbuf[cur][0])[tid];        // ds_load_b128
        const float4 tv0 = ((const float4*)&tbuf[cur][0])[tid];
        const float4 sv1 = ((const float4*)&sbuf[cur][0])[tid + TPB];
        const float4 tv1 = ((const float4*)&tbuf[cur][0])[tid + TPB];
#else
        const float4 sv0 = ((const float4*)chan_ptr(S, c))[tid];
        const float4 tv0 = ((const float4*)chan_ptr(T, c))[tid];
        const float4 sv1 = ((const float4*)chan_ptr(S, c))[tid + TPB];
        const float4 tv1 = ((const float4*)chan_ptr(T, c))[tid + TPB];
#endif
        const float se[EPT] = {sv0.x, sv0.y, sv0.z, sv0.w,
                               sv1.x, sv1.y, sv1.z, sv1.w};
        const float te[EPT] = {tv0.x, tv0.y, tv0.z, tv0.w,
                               tv1.x, tv1.y, tv1.z, tv1.w};
        float sTe = 0.f, wTe = 0.f, sSe = 0.f;
        float sTb = 0.f, wTb = 0.f, sSb = 0.f;
#pragma unroll
        for (int j = 0; j < EPT; ++j) {
            const unsigned m = pm[j];
            const float eM = (float)((m >> c) & 1u);
            const float bM = (float)((m >> (16 + c)) & 1u);
            const float eS = se[j] * eM, eT = te[j] * eM;
            const float bS = se[j] * bM, bT = te[j] * bM;
            const float xT = __expf(eT);           // masked-off entries -> exp(0)=1
            sTe += xT; wTe += xT * (eT - eS); sSe += __expf(eS);
            const float yT = __expf(bT);
            sTb += yT; wTb += yT * (bT - bS); sSb += __expf(bS);
        }
        // Deterministic wave32 tree reduce, then cross-wave via LDS.
        float v[6] = {sTe, wTe, sSe, sTb, wTb, sSb};
#pragma unroll
        for (int k = 0; k < 6; ++k)
            for (int off = 16; off > 0; off >>= 1)
                v[k] += __shfl_xor(v[k], off, 32);
        if ((tid & 31) == 0) {
            const int wv = tid >> 5;
#pragma unroll
            for (int k = 0; k < 6; ++k) red[wv][k] = v[k];
        }
        __syncthreads();  // also fences LDS tile reads before buffer reuse
        if (tid == 0) {
            float r[6] = {0.f, 0.f, 0.f, 0.f, 0.f, 0.f};
            for (int wv = 0; wv < TPB / 32; ++wv)
#pragma unroll
                for (int k = 0; k < 6; ++k) r[k] += red[wv][k];
            const int chIdx = b * (NCLS - 1) + (c - 1);
            float* P = partials + ((size_t)chIdx * CHUNKS + chunk) * 6;
#pragma unroll
            for (int k = 0; k < 6; ++k) P[k] = r[k];
        }
    }
}

// ---------------------------------------------------------------------------
// Mask kernel: pack edge/body bits for all classes into one u32 per pixel.
//   body_c = erode_c  (only c == center label, all 4 neighbors in-bounds & equal)
//   edge_c = dilate_c & !erode_c, dilate = union of cross labels; class 0 zeroed.
// ---------------------------------------------------------------------------
__global__ void bpkd_mask_kernel(const int* __restrict__ gt,
                                 unsigned* __restrict__ maskbuf) {
    int idx = blockIdx.x * blockDim.x + threadIdx.x;
    if (idx >= BATCH * HW) return;
    int p = idx & (HW - 1);
    int h = p >> 9;          // / 512
    int w = p & (WDIM - 1);
    int g0 = gt[idx];
    int up = (h > 0)        ? gt[idx - WDIM] : -1;
    int dn = (h < HDIM - 1) ? gt[idx + WDIM] : -1;
    int lf = (w > 0)        ? gt[idx - 1]    : -1;
    int rt = (w < WDIM - 1) ? gt[idx + 1]    : -1;
    unsigned ub = 1u << g0;
    if (up >= 0) ub |= 1u << up;
    if (dn >= 0) ub |= 1u << dn;
    if (lf >= 0) ub |= 1u << lf;
    if (rt >= 0) ub |= 1u << rt;
    bool ero = (up == g0) & (dn == g0) & (lf == g0) & (rt == g0);
    unsigned body = (ero && g0 > 0) ? (1u << g0) : 0u;
    unsigned edge = (ub & 0x3FFEu) & ~body;   // drop class 0, drop body bit
    maskbuf[idx] = edge | (body << 16);
}

// ---------------------------------------------------------------------------
// Final kernel: merge CHUNKS partials per (channel, masktype), KL = wT/sT +
// log(sS) - log(sT); deterministic serial combine into the two scalars.
// loss = W * tau^2 * sum_{b,c} KL / (C * B),  C*B = 56,  tau = 1.
// ---------------------------------------------------------------------------
__global__ void bpkd_final_kernel(const float* __restrict__ partials,
                                  float* __restrict__ out) {
    __shared__ float kl[2 * NCH_USED];
    const int t = threadIdx.x;
    if (t < 2 * NCH_USED) {
        const int mt = t / NCH_USED;   // 0 = edge, 1 = body
        const int ch = t % NCH_USED;
        const float* P = partials + (size_t)ch * CHUNKS * 6 + mt * 3;
        float sT = 0.f, wT = 0.f, sS = 0.f;
        for (int k = 0; k < CHUNKS; ++k) {
            sT += P[(size_t)k * 6 + 0];
            wT += P[(size_t)k * 6 + 1];
            sS += P[(size_t)k * 6 + 2];
        }
        kl[t] = wT / sT + logf(sS) - logf(sT);
    }
    __syncthreads();
    if (t == 0) {
        float sE = 0.f, sB = 0.f;
        for (int ch = 0; ch < NCH_USED; ++ch) {
            sE += kl[ch];
            sB += kl[NCH_USED + ch];
        }
        out[0] = 500.0f * sE / 56.0f;   // EDGE_W / (C*B)
        out[1] = 200.0f * sB / 56.0f;   // BODY_W / (C*B)
    }
}

// ---------------------------------------------------------------------------
extern "C" void kernel_launch(void* const* d_in, const int* in_sizes, int n_in,
                              void* d_out, int out_size, void* d_ws, size_t ws_size,
                              hipStream_t stream) {
    (void)in_sizes; (void)n_in; (void)out_size; (void)ws_size;
    const float* S  = (const float*)d_in[0];
    const float* T  = (const float*)d_in[1];
    const int*   gt = (const int*)d_in[2];

    // Workspace layout: [packed masks: B*HW u32 = 4 MB][partials: 52*128*6 f32]
    unsigned* maskbuf  = (unsigned*)d_ws;
    float*    partials = (float*)((char*)d_ws + (size_t)BATCH * HW * sizeof(unsigned));

    bpkd_mask_kernel<<<(BATCH * HW + TPB - 1) / TPB, TPB, 0, stream>>>(gt, maskbuf);
    bpkd_main_kernel<<<BATCH * CHUNKS, TPB, 0, stream>>>(S, T, maskbuf, partials);
    bpkd_final_kernel<<<1, 128, 0, stream>>>(partials, (float*)d_out);
}